// MPNN_3315714752875
// MI455X (gfx1250) — compile-verified
//
#include <hip/hip_runtime.h>

#define HDIM 128
#define GNUM 256

typedef float v2f __attribute__((ext_vector_type(2)));
typedef float v8f __attribute__((ext_vector_type(8)));

static __device__ __forceinline__ v8f wmma4(v2f a, v2f b, v8f c) {
  // D = A(16x4 f32) x B(4x16 f32) + C(16x16 f32), full fp32 tensor path
  return __builtin_amdgcn_wmma_f32_16x16x4_f32(false, a, false, b, (short)0, c,
                                               false, false);
}

// Per-lane base pointer for A (16x4 f32: lanes 0-15 -> K=0,1 ; lanes 16-31 -> K=2,3).
// Row index clamped (not branched) so every load is unconditional & batchable;
// out-of-range rows are discarded by the store-side guard.
static __device__ __forceinline__ const float* a_base(const float* __restrict__ X,
                                                      int row0, int lane, int n) {
  int r = row0 + (lane & 15);
  r = r < n ? r : (n - 1);
  return X + (size_t)r * HDIM + ((lane >> 4) << 1);
}

// B[k][col] = W[col][k], W row-major [128, ldw]; mirrored lane/K split
static __device__ __forceinline__ const float* b_base(const float* __restrict__ W,
                                                      int col0, int lane, int ldw) {
  int c = col0 + (lane & 15);
  return W + (size_t)c * ldw + ((lane >> 4) << 1);
}

static __device__ __forceinline__ v2f ld2(const float* p) {
  return *(const v2f*)p;  // 8B-aligned: global_load_b64
}

// ---------------- layer-0 node embedding: [N,12] -> [N,128] (tiny K, VALU) --------
__global__ __launch_bounds__(128) void k_node12(const float* __restrict__ x,
                                                const float* __restrict__ W,
                                                const float* __restrict__ b,
                                                float* __restrict__ h, int n) {
  __shared__ float xr[12];
  int v = blockIdx.x;
  if (v >= n) return;
  if (threadIdx.x < 12) xr[threadIdx.x] = x[(size_t)v * 12 + threadIdx.x];
  __syncthreads();
  int o = threadIdx.x;
  float acc = b[o];
#pragma unroll
  for (int k = 0; k < 12; ++k) acc += xr[k] * W[o * 12 + k];
  h[(size_t)v * HDIM + o] = acc;
}

// ---------------- generic Y = X @ W^T + b  (W [128,128], WMMA f32) ----------------
__global__ __launch_bounds__(256) void k_gemm_bias(const float* __restrict__ X,
                                                   const float* __restrict__ W,
                                                   const float* __restrict__ b,
                                                   float* __restrict__ Y, int n) {
  int lane = threadIdx.x & 31;
  int wave = threadIdx.x >> 5;  // 8 waves = 8 col tiles
  int row0 = blockIdx.x * 16;
  int col0 = wave * 16;
  if (row0 >= n) return;
  const float* Ap = a_base(X, row0, lane, n);
  const float* Bp = b_base(W, col0, lane, HDIM);
  v8f acc = {};
#pragma unroll
  for (int k0 = 0; k0 < HDIM; k0 += 4) {
    acc = wmma4(ld2(Ap + k0), ld2(Bp + k0), acc);
  }
  int col = col0 + (lane & 15);
  int rbase = row0 + ((lane >> 4) << 3);
  float bias = b[col];
#pragma unroll
  for (int r = 0; r < 8; ++r) {
    int row = rbase + r;
    if (row < n) Y[(size_t)row * HDIM + col] = acc[r] + bias;
  }
}

// ------------- deg / edge-attr-sum precompute (once; src/dst fixed) ---------------
__global__ __launch_bounds__(256) void k_init_degT(float* __restrict__ deg,
                                                   float* __restrict__ T, int n) {
  int v = blockIdx.x * 256 + threadIdx.x;
  if (v >= n) return;
  deg[v] = 1.0f;  // self-loop
  T[v * 3 + 0] = 0.0f; T[v * 3 + 1] = 0.0f; T[v * 3 + 2] = 0.0f;
}

__global__ __launch_bounds__(256) void k_degT(const int* __restrict__ ei,
                                              const float* __restrict__ ea,
                                              float* __restrict__ deg,
                                              float* __restrict__ T, int E) {
  int e = blockIdx.x * 256 + threadIdx.x;
  if (e >= E) return;
  int d = ei[E + e];  // dst row
  atomicAdd(&deg[d], 1.0f);
  atomicAdd(&T[d * 3 + 0], ea[(size_t)e * 3 + 0]);
  atomicAdd(&T[d * 3 + 1], ea[(size_t)e * 3 + 1]);
  atomicAdd(&T[d * 3 + 2], ea[(size_t)e * 3 + 2]);
}

// ------------- fold edge-linear through msg-linear:  Me[128,3], cvec[128] ---------
__global__ __launch_bounds__(128) void k_fold(const float* __restrict__ msgW,
                                              const float* __restrict__ edgeW,
                                              const float* __restrict__ edgeB,
                                              const float* __restrict__ msgB,
                                              float* __restrict__ Me,
                                              float* __restrict__ cvec) {
  int o = threadIdx.x;
  float cv = msgB[o], m0 = 0.0f, m1 = 0.0f, m2 = 0.0f;
  for (int k = 0; k < HDIM; ++k) {
    float w = msgW[(size_t)o * 384 + 128 + k];  // edge slice of W_msg
    cv += w * edgeB[k];
    m0 += w * edgeW[k * 3 + 0];
    m1 += w * edgeW[k * 3 + 1];
    m2 += w * edgeW[k * 3 + 2];
  }
  Me[o * 3 + 0] = m0; Me[o * 3 + 1] = m1; Me[o * 3 + 2] = m2;
  cvec[o] = cv;
}

// ------------- S init (self-loop: S = h), then scatter-add of neighbors ----------
__global__ __launch_bounds__(256) void k_copy4(const float* __restrict__ src,
                                               float* __restrict__ dst, long total4) {
  long i = (long)blockIdx.x * 256 + threadIdx.x;
  long stride = (long)gridDim.x * 256;
  const float4* s = (const float4*)src;
  float4* d = (float4*)dst;
  for (; i < total4; i += stride) d[i] = s[i];
}

__global__ __launch_bounds__(256) void k_scatter(const int* __restrict__ ei,
                                                 const float* __restrict__ h,
                                                 float* __restrict__ S, int E) {
  int lane = threadIdx.x & 31;
  int wave = threadIdx.x >> 5;
  int e = blockIdx.x * 8 + wave;
  if (e >= E) return;
  int s = ei[e];
  int d = ei[E + e];
  float4 v = ((const float4*)(h + (size_t)s * HDIM))[lane];  // 32 lanes x 4 = 128
  float* Sd = S + (size_t)d * HDIM + lane * 4;
  atomicAdd(Sd + 0, v.x);
  atomicAdd(Sd + 1, v.y);
  atomicAdd(Sd + 2, v.z);
  atomicAdd(Sd + 3, v.w);
}

// ------- aggr = deg*(h@Wmd^T + cvec) + T@Me^T + S@Wms^T   (dual WMMA, fused) -----
__global__ __launch_bounds__(256) void k_msg_aggr(const float* __restrict__ h,
                                                  const float* __restrict__ S,
                                                  const float* __restrict__ msgW,
                                                  const float* __restrict__ Me,
                                                  const float* __restrict__ cvec,
                                                  const float* __restrict__ deg,
                                                  const float* __restrict__ T,
                                                  float* __restrict__ aggr, int n) {
  int lane = threadIdx.x & 31;
  int wave = threadIdx.x >> 5;
  int row0 = blockIdx.x * 16;
  int col0 = wave * 16;
  if (row0 >= n) return;
  const float* Ah = a_base(h, row0, lane, n);
  const float* As = a_base(S, row0, lane, n);
  const float* Bd = b_base(msgW, col0, lane, 384);        // dst slice  (cols 0..127)
  const float* Bs = b_base(msgW + 256, col0, lane, 384);  // src slice  (cols 256..383)
  v8f ah = {}, as = {};
#pragma unroll
  for (int k0 = 0; k0 < HDIM; k0 += 4) {
    ah = wmma4(ld2(Ah + k0), ld2(Bd + k0), ah);
    as = wmma4(ld2(As + k0), ld2(Bs + k0), as);
  }
  int col = col0 + (lane & 15);
  int rbase = row0 + ((lane >> 4) << 3);
  float cv = cvec[col];
  float m0 = Me[col * 3 + 0], m1 = Me[col * 3 + 1], m2 = Me[col * 3 + 2];
#pragma unroll
  for (int r = 0; r < 8; ++r) {
    int row = rbase + r;
    int rc = row < n ? row : (n - 1);  // clamp loads, guard store
    float dg = deg[rc];
    float t0 = T[rc * 3 + 0], t1 = T[rc * 3 + 1], t2 = T[rc * 3 + 2];
    float val = dg * (ah[r] + cv) + t0 * m0 + t1 * m1 + t2 * m2 + as[r];
    if (row < n) aggr[(size_t)row * HDIM + col] = val;
  }
}

// ---------- hnew = relu(h@U1^T + aggr@U2^T + b)   (dual WMMA, fused) -------------
__global__ __launch_bounds__(256) void k_update(const float* __restrict__ h,
                                                const float* __restrict__ aggr,
                                                const float* __restrict__ updW,
                                                const float* __restrict__ updB,
                                                float* __restrict__ hn, int n) {
  int lane = threadIdx.x & 31;
  int wave = threadIdx.x >> 5;
  int row0 = blockIdx.x * 16;
  int col0 = wave * 16;
  if (row0 >= n) return;
  const float* A1 = a_base(h, row0, lane, n);
  const float* A2 = a_base(aggr, row0, lane, n);
  const float* B1 = b_base(updW, col0, lane, 256);        // cols   0..127
  const float* B2 = b_base(updW + 128, col0, lane, 256);  // cols 128..255
  v8f a1c = {}, a2c = {};
#pragma unroll
  for (int k0 = 0; k0 < HDIM; k0 += 4) {
    a1c = wmma4(ld2(A1 + k0), ld2(B1 + k0), a1c);
    a2c = wmma4(ld2(A2 + k0), ld2(B2 + k0), a2c);
  }
  int col = col0 + (lane & 15);
  int rbase = row0 + ((lane >> 4) << 3);
  float bias = updB[col];
#pragma unroll
  for (int r = 0; r < 8; ++r) {
    int row = rbase + r;
    if (row < n) {
      float v = a1c[r] + a2c[r] + bias;
      hn[(size_t)row * HDIM + col] = v > 0.0f ? v : 0.0f;
    }
  }
}

// ---------------- global mean pool --------------------------------------------
__global__ __launch_bounds__(256) void k_zero(float* __restrict__ out,
                                              float* __restrict__ cnt) {
  int i = blockIdx.x * 256 + threadIdx.x;
  if (i < GNUM * HDIM) out[i] = 0.0f;
  if (i < GNUM) cnt[i] = 0.0f;
}

__global__ __launch_bounds__(256) void k_pool(const float* __restrict__ h,
                                              const int* __restrict__ batch,
                                              float* __restrict__ out,
                                              float* __restrict__ cnt, int n) {
  int lane = threadIdx.x & 31;
  int wave = threadIdx.x >> 5;
  int v = blockIdx.x * 8 + wave;
  if (v >= n) return;
  int g = batch[v];
  float4 val = ((const float4*)(h + (size_t)v * HDIM))[lane];
  float* o = out + (size_t)g * HDIM + lane * 4;
  atomicAdd(o + 0, val.x);
  atomicAdd(o + 1, val.y);
  atomicAdd(o + 2, val.z);
  atomicAdd(o + 3, val.w);
  if (lane == 0) atomicAdd(&cnt[g], 1.0f);
}

__global__ __launch_bounds__(256) void k_div(float* __restrict__ out,
                                             const float* __restrict__ cnt) {
  int i = blockIdx.x * 256 + threadIdx.x;
  if (i >= GNUM * HDIM) return;
  float c = cnt[i >> 7];
  out[i] = out[i] / (c > 1.0f ? c : 1.0f);
}

// =================================================================================
extern "C" void kernel_launch(void* const* d_in, const int* in_sizes, int n_in,
                              void* d_out, int out_size, void* d_ws, size_t ws_size,
                              hipStream_t stream) {
  const float* x = (const float*)d_in[0];
  const int* ei = (const int*)d_in[1];
  const float* ea = (const float*)d_in[2];
  const int* batch = (const int*)d_in[3];

  int n = in_sizes[0] / 12;  // N nodes
  int E = in_sizes[1] / 2;   // edges (before self-loops)

  // workspace carving (256B aligned)
  char* wp = (char*)d_ws;
  auto carve = [&](size_t bytes) -> void* {
    void* p = (void*)wp;
    wp += (bytes + 255) & ~(size_t)255;
    return p;
  };
  size_t nh = (size_t)n * HDIM * sizeof(float);
  float* hA   = (float*)carve(nh);
  float* hB   = (float*)carve(nh);
  float* S    = (float*)carve(nh);
  float* aggr = (float*)carve(nh);
  float* deg  = (float*)carve((size_t)n * sizeof(float));
  float* T    = (float*)carve((size_t)n * 3 * sizeof(float));
  float* Me   = (float*)carve(3 * HDIM * sizeof(float));
  float* cvec = (float*)carve(HDIM * sizeof(float));
  float* cnt  = (float*)carve(GNUM * sizeof(float));

  float* out = (float*)d_out;

  int mtiles = (n + 15) / 16;

  // graph-structure reductions: identical for all 3 layers, compute once
  k_init_degT<<<(n + 255) / 256, 256, 0, stream>>>(deg, T, n);
  k_degT<<<(E + 255) / 256, 256, 0, stream>>>(ei, ea, deg, T, E);

  const float* hin = nullptr;
  for (int l = 0; l < 3; ++l) {
    const int base = 4 + 8 * l;
    const float* nodeW = (const float*)d_in[base + 0];
    const float* nodeB = (const float*)d_in[base + 1];
    const float* edgeW = (const float*)d_in[base + 2];
    const float* edgeB = (const float*)d_in[base + 3];
    const float* msgW  = (const float*)d_in[base + 4];
    const float* msgB  = (const float*)d_in[base + 5];
    const float* updW  = (const float*)d_in[base + 6];
    const float* updB  = (const float*)d_in[base + 7];

    if (l == 0)
      k_node12<<<n, 128, 0, stream>>>(x, nodeW, nodeB, hA, n);
    else
      k_gemm_bias<<<mtiles, 256, 0, stream>>>(hin, nodeW, nodeB, hA, n);

    k_fold<<<1, 128, 0, stream>>>(msgW, edgeW, edgeB, msgB, Me, cvec);

    k_copy4<<<2048, 256, 0, stream>>>(hA, S, (long)n * HDIM / 4);  // self-loop term
    k_scatter<<<(E + 7) / 8, 256, 0, stream>>>(ei, hA, S, E);

    k_msg_aggr<<<mtiles, 256, 0, stream>>>(hA, S, msgW, Me, cvec, deg, T, aggr, n);
    k_update<<<mtiles, 256, 0, stream>>>(hA, aggr, updW, updB, hB, n);
    hin = hB;
  }

  k_zero<<<(GNUM * HDIM + 255) / 256, 256, 0, stream>>>(out, cnt);
  k_pool<<<(n + 7) / 8, 256, 0, stream>>>(hB, batch, out, cnt, n);
  k_div<<<(GNUM * HDIM + 255) / 256, 256, 0, stream>>>(out, cnt);
}